// Conv_37228776521928
// MI455X (gfx1250) — compile-verified
//
#include <hip/hip_runtime.h>
#include <hip/hip_bf16.h>

typedef __attribute__((ext_vector_type(2))) float v2f;
typedef __attribute__((ext_vector_type(8))) float v8f;
typedef __attribute__((ext_vector_type(4))) int   v4i;

typedef __attribute__((address_space(1))) v4i gv4i;   // global int4
typedef __attribute__((address_space(3))) v4i lv4i;   // LDS int4

#define T_N    30
#define H_N    41
#define WD_N   2048
#define N_SEC  9
#define SEC_LEN 4
#define KH_N   6
#define KW_N   40
#define FM_N   50
#define KTOT   (KH_N * KW_N)      // 240
#define OW_N   (WD_N - KW_N + 1)  // 2009
#define WP_N   (OW_N / 4)         // 502
#define NTILES 126                // ceil(2009/16)
#define SEC_H  9                  // rows per section slab
#define THRESH 23.0f
#define POOLED_ELEMS (N_SEC * T_N * FM_N * WP_N) // 6,777,000

// ---------------------------------------------------------------------------
// Conv + threshold + 4x4 max-pool, fused.  Two blocks per (section, t), each
// owning half of the 126 column tiles.  Implicit GEMM: M = FM (padded to 64),
// N = output columns, K = 240, V_WMMA_F32_16X16X4_F32 (exact f32 math).
// x-slab (9 contiguous rows, 72 KB) staged via async-tensor path into LDS.
// ---------------------------------------------------------------------------
__global__ __launch_bounds__(256) void conv_fire_pool_kernel(
    const float* __restrict__ x,   // (30,1,41,2048)
    const float* __restrict__ W,   // (9,50,1,6,40)
    float* __restrict__ pooled)    // (9,30,50,1,502)
{
    const int blk  = blockIdx.x;        // 0..539
    const int st   = blk >> 1;
    const int csel = blk & 1;           // which half of the column tiles
    const int s = st / T_N;
    const int t = st % T_N;

    // LDS: packed A (weights) 60 KB + x slab 72 KB = 132 KB.
    __shared__ float As[4][KTOT][16];
    __shared__ __align__(16) float xs[SEC_H * WD_N];

    // --- async stage of the contiguous 9-row x slab into LDS -------------
    {
        const float* slab = x + ((size_t)t * H_N + s * SEC_LEN) * WD_N;
        for (int e = threadIdx.x; e < (SEC_H * WD_N) / 4; e += 256) {
            __builtin_amdgcn_global_load_async_to_lds_b128(
                (gv4i*)(slab + e * 4), (lv4i*)(&xs[e * 4]), 0, 0);
        }
    }

    // --- pack weights: As[mtile][k][m] (zero-padded FM 50 -> 64) ---------
    for (int e = threadIdx.x; e < 4 * KTOT * 16; e += 256) {
        const int mt  = e / (KTOT * 16);
        const int rem = e % (KTOT * 16);
        const int k   = rem >> 4;
        const int m   = rem & 15;
        const int fm  = mt * 16 + m;
        float v = 0.0f;
        if (fm < FM_N) {
            const int kh = k / KW_N;
            const int kw = k % KW_N;
            v = W[((s * FM_N + fm) * KH_N + kh) * KW_N + kw];
        }
        As[mt][k][m] = v;
    }

    __builtin_amdgcn_s_wait_asynccnt(0);
    __syncthreads();

    const int wave = threadIdx.x >> 5;
    const int lane = threadIdx.x & 31;
    const int half = lane >> 4;        // 0: K pair {0,1}; 1: K pair {2,3}
    const int lm   = lane & 15;        // M for A, N for B, N for C/D

    const int ntlo = csel * (NTILES / 2);
    const int nthi = ntlo + (NTILES / 2);

    for (int nt = ntlo + wave; nt < nthi; nt += 8) {
        const int owbase = nt * 16;
        unsigned spk[4][8] = {};       // OR of spike ballots across r

        for (int r = 0; r < SEC_LEN; ++r) {
            v8f acc0 = {}; v8f acc1 = {}; v8f acc2 = {}; v8f acc3 = {};

            #pragma unroll 4
            for (int kk = 0; kk < KTOT / 4; ++kk) {
                const int k0  = kk * 4 + half * 2;   // this lane's first K
                const int khA = k0 / KW_N;           // same row for k0, k0+1
                const int kwA = k0 - khA * KW_N;     // even, <= 38
                int cA = owbase + lm + kwA;
                if (cA > WD_N - 2) cA = WD_N - 2;    // only trims ow >= 2009

                const int xoff = (r + khA) * WD_N + cA;
                v2f b;
                b.x = xs[xoff];
                b.y = xs[xoff + 1];

                v2f a0, a1, a2, a3;
                a0.x = As[0][k0][lm];  a0.y = As[0][k0 + 1][lm];
                a1.x = As[1][k0][lm];  a1.y = As[1][k0 + 1][lm];
                a2.x = As[2][k0][lm];  a2.y = As[2][k0 + 1][lm];
                a3.x = As[3][k0][lm];  a3.y = As[3][k0 + 1][lm];

                acc0 = __builtin_amdgcn_wmma_f32_16x16x4_f32(
                    false, a0, false, b, (short)0, acc0, false, false);
                acc1 = __builtin_amdgcn_wmma_f32_16x16x4_f32(
                    false, a1, false, b, (short)0, acc1, false, false);
                acc2 = __builtin_amdgcn_wmma_f32_16x16x4_f32(
                    false, a2, false, b, (short)0, acc2, false, false);
                acc3 = __builtin_amdgcn_wmma_f32_16x16x4_f32(
                    false, a3, false, b, (short)0, acc3, false, false);
            }

            // Threshold + accumulate pool-OR across r.
            // D layout: VGPR v -> lanes 0-15: (M=v, N=lane); 16-31: (M=v+8).
            #pragma unroll
            for (int v = 0; v < 8; ++v) {
                spk[0][v] |= __builtin_amdgcn_ballot_w32(acc0[v] > THRESH);
                spk[1][v] |= __builtin_amdgcn_ballot_w32(acc1[v] > THRESH);
                spk[2][v] |= __builtin_amdgcn_ballot_w32(acc2[v] > THRESH);
                spk[3][v] |= __builtin_amdgcn_ballot_w32(acc3[v] > THRESH);
            }
        }

        // Pool across 4 output columns (bits g*4..g*4+3) and store 0/1.
        if (lane < 8) {
            const int g   = lane & 3;      // pool group within tile
            const int hi  = lane >> 2;     // 0 -> M=v, 1 -> M=v+8
            const int owp = nt * 4 + g;
            if (owp < WP_N) {
                #pragma unroll
                for (int mt = 0; mt < 4; ++mt) {
                    #pragma unroll
                    for (int v = 0; v < 8; ++v) {
                        const int fm = mt * 16 + v + hi * 8;
                        if (fm < FM_N) {
                            const unsigned nib =
                                (spk[mt][v] >> (hi * 16 + g * 4)) & 0xFu;
                            pooled[((size_t)(s * T_N + t) * FM_N + fm) * WP_N + owp] =
                                nib ? 1.0f : 0.0f;
                        }
                    }
                }
            }
        }
    }
}

// ---------------------------------------------------------------------------
// Per-section winner: earliest spike time, ties -> lowest (fm*502+owp).
// score = (T - t_first)*BIG + 1.0  ==>  minimize key = (t_first<<15)|flat.
// ---------------------------------------------------------------------------
__global__ __launch_bounds__(256) void winner_kernel(
    const float* __restrict__ pooled, float* __restrict__ feat)
{
    const int s = blockIdx.x;
    const int nflat = FM_N * WP_N;     // 25100 < 2^15
    unsigned best = 0xFFFFFFFFu;

    for (int flat = threadIdx.x; flat < nflat; flat += 256) {
        const int fm  = flat / WP_N;
        const int owp = flat % WP_N;
        int tf = T_N;
        for (int t = 0; t < T_N; ++t) {
            const float v =
                pooled[((size_t)(s * T_N + t) * FM_N + fm) * WP_N + owp];
            if (v > 0.0f) { tf = t; break; }
        }
        const unsigned key = ((unsigned)tf << 15) | (unsigned)flat;
        best = best < key ? best : key;
    }

    __shared__ unsigned red[256];
    red[threadIdx.x] = best;
    __syncthreads();
    for (int off = 128; off > 0; off >>= 1) {
        if (threadIdx.x < (unsigned)off) {
            const unsigned o = red[threadIdx.x + off];
            if (o < red[threadIdx.x]) red[threadIdx.x] = o;
        }
        __syncthreads();
    }
    if (threadIdx.x == 0) {
        const unsigned k = red[0];
        const int tf   = (int)(k >> 15);
        const int flat = (int)(k & 0x7FFFu);
        feat[s] = (tf >= T_N) ? -1.0f : (float)(flat / WP_N);
    }
}

extern "C" void kernel_launch(void* const* d_in, const int* in_sizes, int n_in,
                              void* d_out, int out_size, void* d_ws, size_t ws_size,
                              hipStream_t stream) {
    (void)in_sizes; (void)n_in; (void)d_ws; (void)ws_size; (void)out_size;
    const float* x = (const float*)d_in[0];
    const float* W = (const float*)d_in[1];
    float* out = (float*)d_out;

    conv_fire_pool_kernel<<<N_SEC * T_N * 2, 256, 0, stream>>>(x, W, out);
    winner_kernel<<<N_SEC, 256, 0, stream>>>(out, out + POOLED_ELEMS);
}